// GATConvTemporalPool_8684423873297
// MI455X (gfx1250) — compile-verified
//
#include <hip/hip_runtime.h>
#include <hip/hip_bf16.h>
#include <math.h>

// Problem constants (match reference)
#define NN   20000
#define EE   480000
#define TT   3
#define INF_ 128
#define CC   64
#define HH   2
#define MAXT 4
#define KK   10000          // ceil(0.5*N)
#define HC   (HH*CC)        // 128
#define TN   (TT*NN)        // 60000
#define TE   (TT*EE)        // 1440000
#define FEATW (TT*CC)       // 192

typedef __attribute__((ext_vector_type(2))) float v2f;
typedef __attribute__((ext_vector_type(8))) float v8f;

// ---------- monotone float<->uint encoding for atomic max ----------
__device__ __forceinline__ unsigned f2ord(float f) {
    unsigned u = __float_as_uint(f);
    return (u & 0x80000000u) ? ~u : (u | 0x80000000u);
}
__device__ __forceinline__ float ord2f(unsigned u) {
    if (u == 0u) return 0.0f;                    // "no edge" sentinel -> 0 (matches isfinite-where)
    return (u & 0x80000000u) ? __uint_as_float(u ^ 0x80000000u)
                             : __uint_as_float(~u);
}
__device__ __forceinline__ float leaky(float a) { return a > 0.0f ? a : 0.2f * a; }

// ================= Kernel 1: h = x @ W via f32 WMMA ==================
// x: [TN, 128], W: [128, 128], h: [TN, 128]
// One wave per 16x16 tile; block = 8 waves = one 16-row slab, 8 col tiles.
__global__ __launch_bounds__(256) void k_gemm_xw(const float* __restrict__ x,
                                                 const float* __restrict__ W,
                                                 float* __restrict__ h) {
    const int wave = threadIdx.x >> 5;
    const int lane = threadIdx.x & 31;
    const int row0 = blockIdx.x * 16;     // 3750 blocks
    const int col0 = wave * 16;           // 8 col tiles
    const int m  = lane & 15;             // M (A) / N (B) index
    const int kl = (lane >> 4) * 2;       // lanes 16..31 hold K+2

    v8f acc = {};
    const float* __restrict__ xrow = x + (size_t)(row0 + m) * INF_;
#pragma unroll
    for (int kb = 0; kb < INF_; kb += 4) {
        v2f a, b;
        a.x = xrow[kb + kl + 0];
        a.y = xrow[kb + kl + 1];
        b.x = W[(kb + kl + 0) * HC + col0 + m];
        b.y = W[(kb + kl + 1) * HC + col0 + m];
        acc = __builtin_amdgcn_wmma_f32_16x16x4_f32(
            /*neg_a=*/false, a, /*neg_b=*/false, b,
            /*c_mod=*/(short)0, acc, /*reuse_a=*/false, /*reuse_b=*/false);
    }
    // C/D layout: VGPR v, lanes 0-15 -> row v, lanes 16-31 -> row v+8; col = lane&15
    const int col   = lane & 15;
    const int rbase = (lane >> 4) * 8;
#pragma unroll
    for (int v = 0; v < 8; ++v)
        h[(size_t)(row0 + rbase + v) * HC + col0 + col] = acc[v];
}

// ===== Kernel 2: per-node attention scalars s = [si0, sj0, si1, sj1] =====
__global__ void k_node_scalars(const float* __restrict__ h,
                               const float* __restrict__ att,   // [H, 2C] = 256 floats
                               float* __restrict__ s) {
    int n = blockIdx.x * blockDim.x + threadIdx.x;
    if (n >= TN) return;
    const float* hr = h + (size_t)n * HC;
    float si0 = 0.f, sj0 = 0.f, si1 = 0.f, sj1 = 0.f;
#pragma unroll 4
    for (int c = 0; c < CC; ++c) {
        si0 += hr[c]      * att[c];
        sj0 += hr[c]      * att[CC + c];
        si1 += hr[CC + c] * att[2 * CC + c];
        sj1 += hr[CC + c] * att[3 * CC + c];
    }
    float* sp = s + (size_t)n * 4;
    sp[0] = si0; sp[1] = sj0; sp[2] = si1; sp[3] = sj1;
}

__device__ __forceinline__ void edge_alpha(const int* __restrict__ ei,
                                           const float* __restrict__ ew,
                                           const float* __restrict__ s,
                                           int t, int e,
                                           int& src, int& dst, float& w,
                                           float& a0, float& a1) {
    src = ei[t * 2 * EE + e];
    dst = ei[t * 2 * EE + EE + e];
    w   = ew[t * EE + e];
    const float* ss = s + (size_t)(t * NN + src) * 4;
    const float* sd = s + (size_t)(t * NN + dst) * 4;
    a0 = leaky(sd[0] + w * ss[1]);
    a1 = leaky(sd[2] + w * ss[3]);
}

// ===== Kernel 3: segment max of alpha over dst (encoded-uint atomicMax) =====
__global__ void k_edge_amax(const int* __restrict__ ei, const float* __restrict__ ew,
                            const float* __restrict__ s, unsigned* __restrict__ amax) {
    int idx = blockIdx.x * blockDim.x + threadIdx.x;
    if (idx >= TE) return;
    int t = idx / EE, e = idx - t * EE, src, dst; float w, a0, a1;
    edge_alpha(ei, ew, s, t, e, src, dst, w, a0, a1);
    unsigned* ap = amax + (size_t)(t * NN + dst) * 2;
    atomicMax(&ap[0], f2ord(a0));
    atomicMax(&ap[1], f2ord(a1));
}

// ===== Kernel 4: softmax denominator =====
__global__ void k_edge_den(const int* __restrict__ ei, const float* __restrict__ ew,
                           const float* __restrict__ s, const unsigned* __restrict__ amax,
                           float* __restrict__ den) {
    int idx = blockIdx.x * blockDim.x + threadIdx.x;
    if (idx >= TE) return;
    int t = idx / EE, e = idx - t * EE, src, dst; float w, a0, a1;
    edge_alpha(ei, ew, s, t, e, src, dst, w, a0, a1);
    const unsigned* ap = amax + (size_t)(t * NN + dst) * 2;
    float* dp = den + (size_t)(t * NN + dst) * 2;
    atomicAdd(&dp[0], expf(a0 - ord2f(ap[0])));
    atomicAdd(&dp[1], expf(a1 - ord2f(ap[1])));
}

// ===== Kernel 5: message aggregation (wave per edge, lane = 4 channels) =====
__global__ __launch_bounds__(256) void k_edge_agg(const int* __restrict__ ei,
                                                  const float* __restrict__ ew,
                                                  const float* __restrict__ s,
                                                  const unsigned* __restrict__ amax,
                                                  const float* __restrict__ den,
                                                  const float* __restrict__ h,
                                                  float* __restrict__ agg) {
    int idx = blockIdx.x * 8 + (threadIdx.x >> 5);   // edge id over T*E
    if (idx >= TE) return;
    const int lane = threadIdx.x & 31;
    int t = idx / EE, e = idx - t * EE, src, dst; float w, a0, a1;
    edge_alpha(ei, ew, s, t, e, src, dst, w, a0, a1);
    const size_t gd = (size_t)(t * NN + dst);
    const unsigned* ap = amax + gd * 2;
    const float*    dp = den  + gd * 2;
    float at0 = expf(a0 - ord2f(ap[0])) / (dp[0] + 1e-16f);
    float at1 = expf(a1 - ord2f(ap[1])) / (dp[1] + 1e-16f);
    float scale = ((lane < 16) ? at0 : at1) * w;     // a * ew (xj weighting)

    const float* __restrict__ hs = h + (size_t)(t * NN + src) * HC + lane * 4;
    float*       __restrict__ ag = agg + gd * HC + lane * 4;
    const float4 hv = *(const float4*)hs;
    atomicAdd(&ag[0], hv.x * scale);
    atomicAdd(&ag[1], hv.y * scale);
    atomicAdd(&ag[2], hv.z * scale);
    atomicAdd(&ag[3], hv.w * scale);
}

// ===== Kernel 6: 1/||pool_p|| =====
__global__ __launch_bounds__(256) void k_pool_norm(const float* __restrict__ p,
                                                   float* __restrict__ inv_norm) {
    __shared__ float red[256];
    float v = p[threadIdx.x];                        // exactly MAXT*C = 256 elements
    red[threadIdx.x] = v * v;
    __syncthreads();
    for (int st = 128; st > 0; st >>= 1) {
        if ((int)threadIdx.x < st) red[threadIdx.x] += red[threadIdx.x + st];
        __syncthreads();
    }
    if (threadIdx.x == 0) inv_norm[0] = rsqrtf(red[0]);
}

// ===== Kernel 7: head-mean + bias -> feat [N,192], score = tanh(feat.p/||p||) =====
__global__ void k_finalize(const float* __restrict__ agg, const float* __restrict__ bias,
                           const float* __restrict__ pool_p, const float* __restrict__ inv_norm,
                           float* __restrict__ feat, float* __restrict__ score) {
    int n = blockIdx.x * blockDim.x + threadIdx.x;
    if (n >= NN) return;
    float dot = 0.f;
    for (int t = 0; t < TT; ++t) {
        const float* ag = agg + (size_t)(t * NN + n) * HC;
        float* fr = feat + (size_t)n * FEATW + t * CC;
#pragma unroll 4
        for (int c = 0; c < CC; ++c) {
            float conv = 0.5f * (ag[c] + ag[CC + c]) + bias[c];
            fr[c] = conv;
            dot += conv * pool_p[t * CC + c];        // padded cols are zero
        }
    }
    score[n] = tanhf(dot * inv_norm[0]);
}

// ===== Kernel 8: exact top-k rank by counting (jax tie-break: lower index wins) =====
__global__ __launch_bounds__(256) void k_rank(const float* __restrict__ score,
                                              int* __restrict__ sel) {
    __shared__ float tile[256];
    const int i = blockIdx.x * 256 + threadIdx.x;
    const float my = (i < NN) ? score[i] : 0.f;
    int rank = 0;
    for (int j0 = 0; j0 < NN; j0 += 256) {
        int j = j0 + (int)threadIdx.x;
        tile[threadIdx.x] = (j < NN) ? score[j] : -INFINITY;
        __syncthreads();
        int lim = (NN - j0 < 256) ? (NN - j0) : 256;
        for (int jj = 0; jj < lim; ++jj) {
            float sv = tile[jj];
            rank += (sv > my) || (sv == my && (j0 + jj) < i);
        }
        __syncthreads();
    }
    if (i < NN && rank < KK) sel[rank] = i;
}

// ===== Kernel 9: emit out[t, r, c] = feat[sel[r], t*C+c] * score[sel[r]] =====
__global__ void k_emit(const float* __restrict__ feat, const float* __restrict__ score,
                       const int* __restrict__ sel, float* __restrict__ out) {
    const int r = blockIdx.x;           // 0..K-1
    const int c = threadIdx.x;          // 0..191
    const int i = sel[r];
    const float v = score[i];
    const int t = c >> 6, cc = c & 63;
    out[((size_t)t * KK + r) * CC + cc] = feat[(size_t)i * FEATW + c] * v;
}

extern "C" void kernel_launch(void* const* d_in, const int* in_sizes, int n_in,
                              void* d_out, int out_size, void* d_ws, size_t ws_size,
                              hipStream_t stream) {
    const float* x      = (const float*)d_in[0];   // [T,N,128]
    const int*   ei     = (const int*)  d_in[1];   // [T,2,E]
    const float* ew     = (const float*)d_in[2];   // [T,E]
    const float* W      = (const float*)d_in[3];   // [128,128]
    const float* att    = (const float*)d_in[4];   // [1,H,2C] = 256
    const float* bias   = (const float*)d_in[5];   // [64]
    const float* pool_p = (const float*)d_in[6];   // [256]
    float* out = (float*)d_out;

    // ---- workspace carve (aligned 256B) ----
    char* base = (char*)d_ws;
    size_t off = 0;
    auto carve = [&](size_t bytes) {
        char* p = base + off;
        off = (off + bytes + 255) & ~(size_t)255;
        return p;
    };
    float*    h        = (float*)   carve((size_t)TN * HC * 4);   // 30.72 MB
    float*    agg      = (float*)   carve((size_t)TN * HC * 4);   // 30.72 MB
    float*    feat     = (float*)   carve((size_t)NN * FEATW * 4);// 15.36 MB
    float*    s        = (float*)   carve((size_t)TN * 4 * 4);
    unsigned* amax     = (unsigned*)carve((size_t)TN * 2 * 4);
    float*    den      = (float*)   carve((size_t)TN * 2 * 4);
    float*    score    = (float*)   carve((size_t)NN * 4);
    int*      sel      = (int*)     carve((size_t)KK * 4);
    float*    inv_norm = (float*)   carve(256);
    (void)ws_size; (void)n_in; (void)in_sizes; (void)out_size;

    // zero accumulators (graph-capturable async memsets)
    hipMemsetAsync(agg,  0, (size_t)TN * HC * 4, stream);
    hipMemsetAsync(amax, 0, (size_t)TN * 2 * 4, stream);   // 0 = "-inf" sentinel
    hipMemsetAsync(den,  0, (size_t)TN * 2 * 4, stream);

    k_pool_norm<<<1, 256, 0, stream>>>(pool_p, inv_norm);
    k_gemm_xw<<<TN / 16, 256, 0, stream>>>(x, W, h);
    k_node_scalars<<<(TN + 255) / 256, 256, 0, stream>>>(h, att, s);
    k_edge_amax<<<(TE + 255) / 256, 256, 0, stream>>>(ei, ew, s, amax);
    k_edge_den <<<(TE + 255) / 256, 256, 0, stream>>>(ei, ew, s, amax, den);
    k_edge_agg <<<(TE + 7) / 8,     256, 0, stream>>>(ei, ew, s, amax, den, h, agg);
    k_finalize <<<(NN + 255) / 256, 256, 0, stream>>>(agg, bias, pool_p, inv_norm, feat, score);
    k_rank     <<<(NN + 255) / 256, 256, 0, stream>>>(score, sel);
    k_emit     <<<KK, FEATW, 0, stream>>>(feat, score, sel, out);
}